// GraphicalBranch_drg_84610855731248
// MI455X (gfx1250) — compile-verified
//
#include <hip/hip_runtime.h>
#include <hip/hip_bf16.h>

typedef __attribute__((ext_vector_type(2))) float v2f;
typedef __attribute__((ext_vector_type(8))) float v8f;

#define NOBJ      8
#define NC2       28
#define BATCH     1024
#define DIM       512
#define MAXREL    10
#define NROWS     (BATCH * NC2)          // 28672
#define NOUT      (BATCH * MAXREL)       // 10240
#define LUTSZ     (BATCH * NOBJ * NOBJ)  // 65536

// ---------------------------------------------------------------------------
// 1) Per-sample sums: S[b][d] = sum_{p<28} x[(b*28+p)*512 + d]
// ---------------------------------------------------------------------------
__global__ void sum_nodes_kernel(const float* __restrict__ x, float* __restrict__ S) {
    int idx = blockIdx.x * blockDim.x + threadIdx.x;     // 1024*512 threads
    int b = idx >> 9;
    int d = idx & (DIM - 1);
    const float* p = x + (size_t)b * NC2 * DIM + d;
    float s = 0.f;
#pragma unroll
    for (int i = 0; i < NC2; ++i) s += p[(size_t)i * DIM];
    S[idx] = s;
}

// ---------------------------------------------------------------------------
// 2) LUT scatter (faithful to reference: zeros + scatter of slicing_tensor)
// ---------------------------------------------------------------------------
__global__ void lut_zero_kernel(int* __restrict__ lut) {
    int i = blockIdx.x * blockDim.x + threadIdx.x;
    if (i < LUTSZ) lut[i] = 0;
}

__global__ void lut_fill_kernel(const int* __restrict__ st, int* __restrict__ lut) {
    int r = blockIdx.x * blockDim.x + threadIdx.x;
    if (r < NROWS) {
        int bb = st[3 * r + 0];
        int ii = st[3 * r + 1];
        int jj = st[3 * r + 2];
        lut[bb * (NOBJ * NOBJ) + ii * NOBJ + jj] = r;
    }
}

// ---------------------------------------------------------------------------
// 3) Resolve gather rows: rows[t] = lut[b*64 + pmin*8 + pmax]
// ---------------------------------------------------------------------------
__global__ void rows_kernel(const int* __restrict__ object_pairs,
                            const int* __restrict__ lut,
                            int* __restrict__ rows) {
    int t = blockIdx.x * blockDim.x + threadIdx.x;
    if (t < NOUT) {
        int b  = t / MAXREL;
        int i0 = object_pairs[2 * t + 0];
        int i1 = object_pairs[2 * t + 1];
        int pmin = i0 < i1 ? i0 : i1;
        int pmax = i0 < i1 ? i1 : i0;
        rows[t] = lut[b * (NOBJ * NOBJ) + pmin * NOBJ + pmax];
    }
}

// ---------------------------------------------------------------------------
// 4) G = S @ W_nbr   [1024,512] x [512,512] fp32 WMMA
//    One wave per 16x16 tile; 4 waves per 128-thread block.
//    Tiles: (1024/16)=64 row tiles x (512/16)=32 col tiles = 2048 -> 512 blocks
// ---------------------------------------------------------------------------
__global__ void gemm_g_kernel(const float* __restrict__ S,
                              const float* __restrict__ Wnbr,
                              float* __restrict__ G) {
    const int lane = threadIdx.x & 31;
    const int wave = threadIdx.x >> 5;
    const int tile = blockIdx.x * 4 + wave;
    const int tileN = tile & 31;          // 32 col tiles
    const int tileM = tile >> 5;          // 64 row tiles
    const int mn = lane & 15;             // A-row within tile == B/D column within tile
    const int h  = lane >> 4;             // k-half selector
    const float* arow = S + (size_t)(tileM * 16 + mn) * DIM;
    const int n0 = tileN * 16;

    v8f c = {};
#pragma unroll 4
    for (int k = 0; k < DIM; k += 4) {
        // A 16x4: comp0=A[m][k+2h], comp1=A[m][k+2h+1]  (contiguous -> b64 load)
        v2f a = *(const v2f*)(arow + k + 2 * h);
        // B 4x16: comp0=W[k+2h][n], comp1=W[k+2h+1][n]
        v2f bv;
        bv.x = Wnbr[(size_t)(k + 2 * h + 0) * DIM + n0 + mn];
        bv.y = Wnbr[(size_t)(k + 2 * h + 1) * DIM + n0 + mn];
        c = __builtin_amdgcn_wmma_f32_16x16x4_f32(
                /*neg_a=*/false, a, /*neg_b=*/false, bv,
                /*c_mod=*/(short)0, c, /*reuse_a=*/false, /*reuse_b=*/false);
    }
#pragma unroll
    for (int v = 0; v < 8; ++v) {
        int orow = tileM * 16 + v + 8 * h;   // C/D: VGPR v -> M = v + 8*(lane>>4)
        G[(size_t)orow * DIM + n0 + mn] = c[v];
    }
}

// ---------------------------------------------------------------------------
// 5) out[t] = relu( x[rows[t]] @ W_self + G[t/10] + bias )
//    Tiles: (10240/16)=640 row tiles x 32 col tiles = 20480 -> 5120 blocks
// ---------------------------------------------------------------------------
__global__ void gemm_out_kernel(const float* __restrict__ x,
                                const float* __restrict__ Wself,
                                const float* __restrict__ G,
                                const float* __restrict__ bias,
                                const int* __restrict__ rows,
                                float* __restrict__ out) {
    const int lane = threadIdx.x & 31;
    const int wave = threadIdx.x >> 5;
    const int tile = blockIdx.x * 4 + wave;
    const int tileN = tile & 31;          // 32 col tiles
    const int tileM = tile >> 5;          // 640 row tiles
    const int mn = lane & 15;
    const int h  = lane >> 4;
    const int srcRow = rows[tileM * 16 + mn];          // gathered A row in x
    const float* arow = x + (size_t)srcRow * DIM;
    const int n0 = tileN * 16;

    v8f c = {};
#pragma unroll 4
    for (int k = 0; k < DIM; k += 4) {
        v2f a = *(const v2f*)(arow + k + 2 * h);
        v2f bv;
        bv.x = Wself[(size_t)(k + 2 * h + 0) * DIM + n0 + mn];
        bv.y = Wself[(size_t)(k + 2 * h + 1) * DIM + n0 + mn];
        c = __builtin_amdgcn_wmma_f32_16x16x4_f32(
                false, a, false, bv, (short)0, c, false, false);
    }

    const float bcol = bias[n0 + mn];
#pragma unroll
    for (int v = 0; v < 8; ++v) {
        int orow = tileM * 16 + v + 8 * h;             // output row t
        int col  = n0 + mn;
        float val = c[v] + G[(size_t)(orow / MAXREL) * DIM + col] + bcol;
        out[(size_t)orow * DIM + col] = fmaxf(val, 0.f);
    }
}

// ---------------------------------------------------------------------------
// Launch
// ---------------------------------------------------------------------------
extern "C" void kernel_launch(void* const* d_in, const int* in_sizes, int n_in,
                              void* d_out, int out_size, void* d_ws, size_t ws_size,
                              hipStream_t stream) {
    const float* x     = (const float*)d_in[0];   // [28672, 512]
    const float* Wself = (const float*)d_in[1];   // [512, 512]
    const float* Wnbr  = (const float*)d_in[2];   // [512, 512]
    const float* bias  = (const float*)d_in[3];   // [512]
    const int*   slice = (const int*)d_in[4];     // [28672, 3]
    const int*   opair = (const int*)d_in[5];     // [1024, 10, 2]
    // d_in[6] num_rel, d_in[7] num_obj: static in reference, unused.

    char* ws = (char*)d_ws;
    float* S   = (float*)(ws);                                   // 2 MB
    float* G   = (float*)(ws + (size_t)BATCH * DIM * 4);         // 2 MB
    int*   lut = (int*)  (ws + (size_t)2 * BATCH * DIM * 4);     // 256 KB
    int*   rows= (int*)  (ws + (size_t)2 * BATCH * DIM * 4 + (size_t)LUTSZ * 4);
    float* out = (float*)d_out;                                  // [10240, 512]

    sum_nodes_kernel<<<(BATCH * DIM) / 256, 256, 0, stream>>>(x, S);
    lut_zero_kernel <<<LUTSZ / 256, 256, 0, stream>>>(lut);
    lut_fill_kernel <<<(NROWS + 255) / 256, 256, 0, stream>>>(slice, lut);
    rows_kernel     <<<(NOUT + 255) / 256, 256, 0, stream>>>(opair, lut, rows);

    // G = S @ W_nbr : 64*32 tiles, 4 waves/block
    gemm_g_kernel  <<<(64 * 32) / 4, 128, 0, stream>>>(S, Wnbr, G);
    // out = relu(gather(x) @ W_self + G + b) : 640*32 tiles
    gemm_out_kernel<<<(640 * 32) / 4, 128, 0, stream>>>(x, Wself, G, bias, rows, out);
}